// GATLayer_57561151701647
// MI455X (gfx1250) — compile-verified
//
#include <hip/hip_runtime.h>
#include <math.h>
#include <limits.h>

#define N_NODES   50000
#define N_EDGES   800000
#define IN_FEATS  128
#define OUT_FEATS 16
#define NUM_HEADS 4
#define ZCOLS     (NUM_HEADS * OUT_FEATS)   // 64
#define ALPHA     0.2f

typedef __attribute__((ext_vector_type(2))) float v2f;
typedef __attribute__((ext_vector_type(8))) float v8f;

// ---------------------------------------------------------------------------
// K1: z = h @ W^T  via V_WMMA_F32_16X16X4_F32 (full fp32, matches reference).
// grid = N/16 blocks, 128 threads (4 waves). Wave w owns C tile:
//   mtile = blockIdx.x (16 nodes), ntile = w (16 out-feats == one head).
// A (16x4 f32) lane layout: row = lane&15, VGPR0/1 hold K = kbase, kbase+1
// with kbase = (lane>>4)*2.  B (4x16) mirrors it: col = lane&15, rows
// kbase, kbase+1.  C (16x16): VGPR r -> M = r + 8*(lane>>4), N = lane&15.
// EXEC is all-ones everywhere (exact grid), as WMMA requires.
// ---------------------------------------------------------------------------
__global__ __launch_bounds__(128) void gat_proj_wmma(
    const float* __restrict__ h, const float* __restrict__ W,
    float* __restrict__ z)
{
  const int wave  = threadIdx.x >> 5;
  const int lane  = threadIdx.x & 31;
  const int mtile = blockIdx.x;
  const int ntile = wave;                    // == head index
  const int lo16  = lane & 15;
  const int kbase = (lane >> 4) * 2;         // 0 or 2

  const float* __restrict__ ha = h + (size_t)(mtile * 16 + lo16) * IN_FEATS + kbase;
  const float* __restrict__ wb = W + (size_t)(ntile * 16 + lo16) * IN_FEATS + kbase;

  v8f c = {};
#pragma unroll 4
  for (int k0 = 0; k0 < IN_FEATS; k0 += 4) {
    v2f a, b;
    a.x = ha[k0]; a.y = ha[k0 + 1];          // h[row][k0+kbase .. +1]
    b.x = wb[k0]; b.y = wb[k0 + 1];          // W[col][k0+kbase .. +1] = B[k][col]
    c = __builtin_amdgcn_wmma_f32_16x16x4_f32(
        /*neg_a=*/false, a, /*neg_b=*/false, b,
        /*c_mod=*/(short)0, c, /*reuse_a=*/false, /*reuse_b=*/false);
  }

  const int rbase = mtile * 16 + ((lane >> 4) << 3);
  float* __restrict__ zp = z + (size_t)rbase * ZCOLS + ntile * 16 + lo16;
#pragma unroll
  for (int r = 0; r < 8; ++r)
    zp[(size_t)r * ZCOLS] = c[r];
}

// ---------------------------------------------------------------------------
// K2: per-(node,head) half-scores: asrc = z[n,h,:].A[h,:F], adst = ...A[h,F:]
//     also seeds the ordered-int segment-max buffer with INT_MIN.
// ---------------------------------------------------------------------------
__global__ void gat_node_scores(const float* __restrict__ z,
                                const float* __restrict__ A,
                                float* __restrict__ asrc,
                                float* __restrict__ adst,
                                int* __restrict__ maxbuf)
{
  int t = blockIdx.x * blockDim.x + threadIdx.x;
  if (t >= N_NODES * NUM_HEADS) return;
  const int n = t >> 2, hh = t & 3;
  const float* __restrict__ zv = z + (size_t)n * ZCOLS + hh * OUT_FEATS;
  const float* __restrict__ As = A + hh * (2 * OUT_FEATS);
  float s = 0.f, d = 0.f;
#pragma unroll
  for (int f = 0; f < OUT_FEATS; ++f) {
    s += zv[f] * As[f];
    d += zv[f] * As[OUT_FEATS + f];
  }
  asrc[t] = s;
  adst[t] = d;
  maxbuf[t] = INT_MIN;
}

// monotone float -> signed-int encoding (order-preserving), for atomicMax
__device__ __forceinline__ int f32_ord(float v) {
  int i = __float_as_int(v);
  return i >= 0 ? i : (i ^ 0x7FFFFFFF);
}

// ---------------------------------------------------------------------------
// K3: per-(edge,head) logit: leaky_relu(asrc[src] + adst[dst]); store and
//     segment-max into dst via ordered-int atomicMax.
// ---------------------------------------------------------------------------
__global__ void gat_edge_logits(const int* __restrict__ src,
                                const int* __restrict__ dst,
                                const float* __restrict__ asrc,
                                const float* __restrict__ adst,
                                float* __restrict__ eval,
                                int* __restrict__ maxbuf)
{
  int t = blockIdx.x * blockDim.x + threadIdx.x;
  if (t >= N_EDGES * NUM_HEADS) return;
  const int e = t >> 2, hh = t & 3;
  const int s = src[e], d = dst[e];
  float v = asrc[s * NUM_HEADS + hh] + adst[d * NUM_HEADS + hh];
  v = v > 0.f ? v : ALPHA * v;              // leaky relu
  eval[t] = v;
  atomicMax(&maxbuf[d * NUM_HEADS + hh], f32_ord(v));
}

// ---------------------------------------------------------------------------
// K4: decode segment max (isfinite guard like reference), zero denom and out.
// ---------------------------------------------------------------------------
__global__ void gat_prep(const int* __restrict__ maxbuf,
                         float* __restrict__ m,
                         float* __restrict__ denom,
                         float* __restrict__ out)
{
  int t = blockIdx.x * blockDim.x + threadIdx.x;
  if (t >= N_NODES * NUM_HEADS) return;
  const int iv = maxbuf[t];
  float mv = 0.f;
  if (iv != INT_MIN) {
    int j = iv >= 0 ? iv : (iv ^ 0x7FFFFFFF);
    mv = __int_as_float(j);
    if (!isfinite(mv)) mv = 0.f;            // nodes w/o in-edges -> 0
  }
  m[t] = mv;
  denom[t] = 0.f;
  float* __restrict__ op = out + (size_t)(t >> 2) * ZCOLS + (t & 3) * OUT_FEATS;
#pragma unroll
  for (int f = 0; f < OUT_FEATS; ++f) op[f] = 0.f;
}

// ---------------------------------------------------------------------------
// K5: per-(edge,head,feat) scatter: out[dst] += exp(e - m[dst]) * z[src].
// 16 consecutive lanes share one (edge,head): coalesced z reads, consecutive
// atomic targets. lane f==0 also accumulates the softmax denominator.
// grid is exact: E*64 = 200000 * 256.
// ---------------------------------------------------------------------------
__global__ __launch_bounds__(256) void gat_edge_accum(
    const int* __restrict__ src, const int* __restrict__ dst,
    const float* __restrict__ eval, const float* __restrict__ m,
    const float* __restrict__ z,
    float* __restrict__ out, float* __restrict__ denom)
{
  const long long t = (long long)blockIdx.x * 256 + threadIdx.x;
  const int f  = (int)(t & 15);
  const int hh = (int)((t >> 4) & 3);
  const int e  = (int)(t >> 6);
  const int s = src[e], d = dst[e];
  const float ex = expf(eval[e * NUM_HEADS + hh] - m[d * NUM_HEADS + hh]);
  atomicAdd(&out[(size_t)d * ZCOLS + hh * OUT_FEATS + f],
            ex * z[(size_t)s * ZCOLS + hh * OUT_FEATS + f]);
  if (f == 0) atomicAdd(&denom[d * NUM_HEADS + hh], ex);
}

// ---------------------------------------------------------------------------
// K6: out /= max(denom, 1e-12), in place.  grid exact: N*64 = 12500*256.
// ---------------------------------------------------------------------------
__global__ __launch_bounds__(256) void gat_finalize(float* __restrict__ out,
                                                    const float* __restrict__ denom)
{
  const int t = blockIdx.x * 256 + threadIdx.x;
  const int n = t >> 6, j = t & 63;
  out[t] /= fmaxf(denom[n * NUM_HEADS + (j >> 4)], 1e-12f);
}

extern "C" void kernel_launch(void* const* d_in, const int* in_sizes, int n_in,
                              void* d_out, int out_size, void* d_ws, size_t ws_size,
                              hipStream_t stream) {
  const float* h   = (const float*)d_in[0];
  const int*   src = (const int*)d_in[1];
  const int*   dst = (const int*)d_in[2];
  const float* W   = (const float*)d_in[3];
  const float* A   = (const float*)d_in[4];
  float* out = (float*)d_out;

  // workspace layout (floats): z | asrc | adst | m | denom | eval | maxbuf(int)
  float* z     = (float*)d_ws;                         // N*64
  float* asrc  = z     + (size_t)N_NODES * ZCOLS;      // N*H
  float* adst  = asrc  + (size_t)N_NODES * NUM_HEADS;
  float* m     = adst  + (size_t)N_NODES * NUM_HEADS;
  float* denom = m     + (size_t)N_NODES * NUM_HEADS;
  float* eval  = denom + (size_t)N_NODES * NUM_HEADS;  // E*H
  int*   maxbuf = (int*)(eval + (size_t)N_EDGES * NUM_HEADS);

  const int NH = N_NODES * NUM_HEADS;    // 200000
  const int EH = N_EDGES * NUM_HEADS;    // 3200000

  gat_proj_wmma<<<N_NODES / 16, 128, 0, stream>>>(h, W, z);
  gat_node_scores<<<(NH + 255) / 256, 256, 0, stream>>>(z, A, asrc, adst, maxbuf);
  gat_edge_logits<<<(EH + 255) / 256, 256, 0, stream>>>(src, dst, asrc, adst, eval, maxbuf);
  gat_prep<<<(NH + 255) / 256, 256, 0, stream>>>(maxbuf, m, denom, out);
  gat_edge_accum<<<(N_EDGES * 64) / 256, 256, 0, stream>>>(src, dst, eval, m, z, out, denom);
  gat_finalize<<<(N_NODES * ZCOLS) / 256, 256, 0, stream>>>(out, denom);
}